// BoxConv2d_32435593020300
// MI455X (gfx1250) — compile-verified
//
#include <hip/hip_runtime.h>
#include <stdint.h>

// ---------------------------------------------------------------------------
// BoxConv2d fused kernel for MI455X (gfx1250).
//
// Strategy (memory-roofline driven, ~84 MB HBM traffic => ~3.6 us @ 23.3TB/s):
//   * one workgroup per (batch, channel, filter-pair)
//   * TDM (tensor_load_to_lds) DMAs the 128x128 input plane asynchronously
//     into LDS with hardware row padding (pad 4 DWORDs per 128 DWORDs) so it
//     lands directly in the interior of a pitch-132, 129-row integral buffer
//   * wave32 shuffle-scan builds the 2D integral image in LDS (66.5KB --
//     only possible thanks to CDNA5's 320KB LDS/WGP)
//   * each output pixel = 16 LDS taps (4 bilinear corners) + ~24 FMAs
// ---------------------------------------------------------------------------

#define IN_PLANES   32
#define NUM_FILTERS 4
#define BATCH       8
#define HH          128
#define WW          128
#define K_TOT       (IN_PLANES * NUM_FILTERS)
#define PITCH       132                      // 129 cols padded; 132 % 64 == 4
#define LDS_BYTES   ((HH + 1) * PITCH * 4)   // 129*132*4 = 68112 B

typedef __attribute__((ext_vector_type(4))) unsigned int u32x4;
typedef __attribute__((ext_vector_type(8))) int          i32x8;
typedef __attribute__((ext_vector_type(4))) int          i32x4;

__device__ __forceinline__ void interp_idx(float coord, int& ci, float& fr) {
    // clip(coord, 0, 128); ci = min(floor, 127); fr = frac  (matches reference)
    float cc = fminf(fmaxf(coord, 0.0f), 128.0f);
    float fl = floorf(cc);
    int   i  = (int)fl;
    if (i > 127) i = 127;
    ci = i;
    fr = cc - (float)i;
}

__global__ __launch_bounds__(256) void
boxconv2d_kernel(const float* __restrict__ input,
                 const float* __restrict__ x_min, const float* __restrict__ x_max,
                 const float* __restrict__ y_min, const float* __restrict__ y_max,
                 float* __restrict__ out)
{
    extern __shared__ float Ip[];            // [129][PITCH] integral image

    const int tid  = threadIdx.x;
    const int lane = tid & 31;
    const int wv   = tid >> 5;               // 8 waves of 32 (wave32)

    const int fh = blockIdx.x & 1;           // which filter pair
    const int bc = blockIdx.x >> 1;
    const int b  = bc / IN_PLANES;
    const int c  = bc % IN_PLANES;

    // ---- Phase 1: issue TDM async copy of input plane -> LDS interior ----
    if (wv == 0) {
        uint64_t ga = (uint64_t)(uintptr_t)(input +
                        (size_t)(b * IN_PLANES + c) * (size_t)(HH * WW));
        uint32_t lds0 = (uint32_t)(uintptr_t)(Ip + PITCH + 1); // I[1][1]

        u32x4 g0;
        g0[0] = 1u;                                        // count=1, user mode
        g0[1] = lds0;                                      // lds_addr (bytes)
        g0[2] = (uint32_t)ga;                              // global_addr[31:0]
        g0[3] = (uint32_t)((ga >> 32) & 0x1FFFFFFu)        // global_addr[56:32]
              | (2u << 30);                                // type = 2 (image)

        i32x8 g1;
        g1[0] = (int)((2u << 16)      // data_size = 4 bytes
                    | (1u << 20)      // pad_enable
                    | (6u << 22)      // pad_interval code 6 = every 128 DWORDs
                    | (3u << 25));    // pad_amount  code 3 = 4 DWORDs -> pitch 132
        g1[1] = (int)(128u << 16);    // tensor_dim0 = 128
        g1[2] = (int)(128u << 16);    // tensor_dim1 = 128
        g1[3] = (int)(128u << 16);    // tile_dim0   = 128
        g1[4] = (int)128;             // tile_dim1   = 128, tile_dim2 = 0
        g1[5] = (int)128;             // tensor_dim0_stride = 128
        g1[6] = (int)(16384u << 16);  // tensor_dim1_stride lo (unused for 2D)
        g1[7] = 0;

        i32x4 gz = {0, 0, 0, 0};
#if __has_include(<hip/amd_detail/amd_gfx1250_TDM.h>)
        i32x8 gz8 = {0, 0, 0, 0, 0, 0, 0, 0};
        __builtin_amdgcn_tensor_load_to_lds(g0, g1, gz, gz, gz8, 0);
#else
        __builtin_amdgcn_tensor_load_to_lds(g0, g1, gz, gz, 0);
#endif
    }

    // Zero border row 0 and column 0 while the DMA is in flight.
    for (int i = tid; i <= HH; i += 256) {
        Ip[i]         = 0.0f;   // row 0
        Ip[i * PITCH] = 0.0f;   // column 0
    }

    if (wv == 0) __builtin_amdgcn_s_wait_tensorcnt(0);
    __syncthreads();

    // ---- Phase 2: row-wise inclusive scan (wave32 Kogge-Stone shuffles) ----
    for (int r = 1 + wv; r <= HH; r += 8) {
        float* row = Ip + r * PITCH;
        float carry = 0.0f;
#pragma unroll
        for (int ch = 0; ch < 4; ++ch) {
            int s = 1 + ch * 32 + lane;
            float v = row[s];
#pragma unroll
            for (int off = 1; off < 32; off <<= 1) {
                float n = __shfl_up(v, off, 32);
                if (lane >= off) v += n;
            }
            v += carry;
            row[s] = v;
            carry = __shfl(v, 31, 32);
        }
    }
    __syncthreads();

    // ---- Phase 3: column-wise inclusive scan ----
    for (int s = 1 + wv; s <= WW; s += 8) {
        float carry = 0.0f;
#pragma unroll
        for (int ch = 0; ch < 4; ++ch) {
            int r = 1 + ch * 32 + lane;
            float v = Ip[r * PITCH + s];
#pragma unroll
            for (int off = 1; off < 32; off <<= 1) {
                float n = __shfl_up(v, off, 32);
                if (lane >= off) v += n;
            }
            v += carry;
            Ip[r * PITCH + s] = v;
            carry = __shfl(v, 31, 32);
        }
    }
    __syncthreads();

    // ---- Phase 4: evaluate 2 filters from the LDS integral image ----
#pragma unroll
    for (int ff = 0; ff < 2; ++ff) {
        const int   f   = fh * 2 + ff;
        const int   k   = c * NUM_FILTERS + f;
        const float xmn = x_min[c * NUM_FILTERS + f];
        const float xmx = x_max[c * NUM_FILTERS + f];
        const float ymn = y_min[c * NUM_FILTERS + f];
        const float ymx = y_max[c * NUM_FILTERS + f];
        float* ob = out + (size_t)(b * K_TOT + k) * (size_t)(HH * WW);

        for (int i = tid; i < HH * WW; i += 256) {
            const int x = i >> 7;
            const int y = i & (WW - 1);

            int ri0, ri1, sj0, sj1;
            float fa, fb, gva, gvb;
            interp_idx((float)x + xmn,        ri0, fa);
            interp_idx((float)x + xmx + 1.0f, ri1, fb);
            interp_idx((float)y + ymn,        sj0, gva);
            interp_idx((float)y + ymx + 1.0f, sj1, gvb);

            const float* r0a = Ip + ri0 * PITCH;
            const float* r0b = r0a + PITCH;
            const float* r1a = Ip + ri1 * PITCH;
            const float* r1b = r1a + PITCH;

            // T(row-pair, fr, s) = row-interpolated integral at column s
            float t1s1  = r1a[sj1]     + fb * (r1b[sj1]     - r1a[sj1]);
            float t1s1p = r1a[sj1 + 1] + fb * (r1b[sj1 + 1] - r1a[sj1 + 1]);
            float t1s0  = r1a[sj0]     + fb * (r1b[sj0]     - r1a[sj0]);
            float t1s0p = r1a[sj0 + 1] + fb * (r1b[sj0 + 1] - r1a[sj0 + 1]);
            float t0s1  = r0a[sj1]     + fa * (r0b[sj1]     - r0a[sj1]);
            float t0s1p = r0a[sj1 + 1] + fa * (r0b[sj1 + 1] - r0a[sj1 + 1]);
            float t0s0  = r0a[sj0]     + fa * (r0b[sj0]     - r0a[sj0]);
            float t0s0p = r0a[sj0 + 1] + fa * (r0b[sj0 + 1] - r0a[sj0 + 1]);

            float top = (t1s1 + gvb * (t1s1p - t1s1)) - (t1s0 + gva * (t1s0p - t1s0));
            float bot = (t0s1 + gvb * (t0s1p - t0s1)) - (t0s0 + gva * (t0s0p - t0s0));

            ob[i] = top - bot;
        }
    }
}

extern "C" void kernel_launch(void* const* d_in, const int* in_sizes, int n_in,
                              void* d_out, int out_size, void* d_ws, size_t ws_size,
                              hipStream_t stream) {
    (void)in_sizes; (void)n_in; (void)out_size; (void)d_ws; (void)ws_size;
    const float* input = (const float*)d_in[0];
    const float* xmn   = (const float*)d_in[1];
    const float* xmx   = (const float*)d_in[2];
    const float* ymn   = (const float*)d_in[3];
    const float* ymx   = (const float*)d_in[4];
    float* out = (float*)d_out;

    dim3 grid(BATCH * IN_PLANES * 2);   // 512 blocks: (b, c, filter-pair)
    dim3 block(256);                    // 8 wave32s
    boxconv2d_kernel<<<grid, block, LDS_BYTES, stream>>>(input, xmn, xmx, ymn, ymx, out);
}